// RegionLoss_Ava_43456479101511
// MI455X (gfx1250) — compile-verified
//
#include <hip/hip_runtime.h>
#include <hip/hip_bf16.h>

#define NAx 5
#define NCx 80
#define NHx 28
#define NWx 28
#define TTx 50
#define NPOSE 14
#define EPSC 1e-4f
#define CELLS (NAx * NHx * NWx)   // 3920

__constant__ float c_aw[NAx] = {0.70458f, 1.26654f, 1.59382f, 2.30548f, 3.52332f};
__constant__ float c_ah[NAx] = {1.18803f, 2.55121f, 4.08321f, 4.94180f, 5.91979f};

typedef __attribute__((ext_vector_type(2))) float v2f;
typedef __attribute__((ext_vector_type(8))) float v8f;

__device__ __forceinline__ float iou_cwh(float ax, float ay, float aw, float ah,
                                         float bx, float by, float bw, float bh) {
  float mx = fminf(ax - aw * 0.5f, bx - bw * 0.5f);
  float Mx = fmaxf(ax + aw * 0.5f, bx + bw * 0.5f);
  float my = fminf(ay - ah * 0.5f, by - bh * 0.5f);
  float My = fmaxf(ay + ah * 0.5f, by + bh * 0.5f);
  float cw = aw + bw - (Mx - mx);
  float ch = ah + bh - (My - my);
  float carea = (cw <= 0.0f || ch <= 0.0f) ? 0.0f : cw * ch;
  float uarea = aw * ah + bw * bh - carea;
  return carea / fmaxf(uarea, 1e-12f);
}

__device__ __forceinline__ float sigmoidf_(float z) {
  return 1.0f / (1.0f + expf(-z));
}

// Wave-level sum of one float per lane using the CDNA5 f32 WMMA:
// A is 16x4 (2 VGPRs/lane); set A-vgpr0 = v, A-vgpr1 = 0, B = all-ones 4x16.
// D[m][n] = sum_k A[m][k] => row sums. Lane L<16 holds rows 0..7 across its
// 8 D VGPRs, lane L>=16 holds rows 8..15; one xor-16 shuffle finishes the sum.
__device__ __forceinline__ float wave_sum_wmma(float v) {
  v2f a; a[0] = v;    a[1] = 0.0f;
  v2f b; b[0] = 1.0f; b[1] = 1.0f;
  v8f c = {0.0f, 0.0f, 0.0f, 0.0f, 0.0f, 0.0f, 0.0f, 0.0f};
  c = __builtin_amdgcn_wmma_f32_16x16x4_f32(false, a, false, b, (short)0, c,
                                            false, false);
  float s = c[0] + c[1] + c[2] + c[3] + c[4] + c[5] + c[6] + c[7];
  s += __shfl_xor(s, 16, 32);
  return s;  // uniform across the wave
}

__global__ void init_kernel(double* acc) {
  if (threadIdx.x < 3) acc[threadIdx.x] = 0.0;
}

// One thread per (b, t): precompute target record.
// Record (12 floats): gx gy gw gh valid cellidx tx ty tw th 0 0
__global__ void targets_kernel(const float* __restrict__ tb, int nT,
                               float* __restrict__ tabg) {
  int idx = blockIdx.x * blockDim.x + threadIdx.x;
  if (idx >= nT) return;
  const float* p = tb + (size_t)idx * 4;
  float gx = p[0] * (float)NWx, gy = p[1] * (float)NHx;
  float gw = p[2] * (float)NWx, gh = p[3] * (float)NHx;
  float* e = tabg + (size_t)idx * 12;
  if (!(p[2] > 0.0f)) {  // invalid target
#pragma unroll
    for (int k = 0; k < 12; ++k) e[k] = 0.0f;
    e[5] = -1.0f;  // never matches a cell
    return;
  }
  // best anchor: argmax of centered-box IoU, first max wins (strict >)
  float best = -1.0f;
  int bn = 0;
#pragma unroll
  for (int n = 0; n < NAx; ++n) {
    float io = iou_cwh(0.0f, 0.0f, gw, gh, 0.0f, 0.0f, c_aw[n], c_ah[n]);
    if (io > best) { best = io; bn = n; }
  }
  int gi = (int)floorf(gx); gi = gi < 0 ? 0 : (gi > NWx - 1 ? NWx - 1 : gi);
  int gj = (int)floorf(gy); gj = gj < 0 ? 0 : (gj > NHx - 1 ? NHx - 1 : gj);
  e[0] = gx; e[1] = gy; e[2] = gw; e[3] = gh;
  e[4] = 1.0f;
  e[5] = (float)(bn * (NHx * NWx) + gj * NWx + gi);
  e[6] = gx - (float)gi;
  e[7] = gy - (float)gj;
  e[8] = logf(gw / c_aw[bn]);
  e[9] = logf(gh / c_ah[bn]);
  e[10] = 0.0f; e[11] = 0.0f;
}

// grid = (ceil(CELLS/256), nB); block = 256 (8 waves). No early returns:
// WMMA requires EXEC all-ones.
__global__ __launch_bounds__(256) void cells_kernel(
    const float* __restrict__ outp, const float* __restrict__ tcls,
    const float* __restrict__ cwt, const float* __restrict__ tabg,
    double* __restrict__ acc) {
  __shared__ float tab[TTx * 12];
  __shared__ float wpos_s[NCx];
  __shared__ float wneg_s[NCx];
  __shared__ float wred[8][4];

  const int tid = threadIdx.x;
  const int b = blockIdx.y;

  for (int k = tid; k < TTx * 12; k += 256)
    tab[k] = tabg[(size_t)b * TTx * 12 + k];
  for (int k = tid; k < NCx; k += 256) {
    float w = cwt[k];
    wpos_s[k] = expf(w);
    wneg_s[k] = expf(1.0f - w);
  }
  __syncthreads();

  const int idx = blockIdx.x * 256 + tid;  // cell within batch: a*784+j*28+i
  const bool active = idx < CELLS;
  float s_main = 0.0f, s_cls = 0.0f, s_n = 0.0f;

  if (active) {
    const int a = idx / (NHx * NWx);
    const int rem = idx % (NHx * NWx);
    const int j = rem / NWx, i = rem % NWx;
    const float* base =
        outp + (size_t)(b * NAx + a) * (5 + NCx) * (NHx * NWx) + rem;

    const float o0 = base[0];
    const float o1 = base[1 * NHx * NWx];
    const float o2 = base[2 * NHx * NWx];
    const float o3 = base[3 * NHx * NWx];
    const float o4 = base[4 * NHx * NWx];

    const float x = sigmoidf_(o0);
    const float y = sigmoidf_(o1);
    const float conf = sigmoidf_(o4);
    const float px = x + (float)i;
    const float py = y + (float)j;
    const float pw = expf(o2) * c_aw[a];
    const float ph = expf(o3) * c_ah[a];

    // silence mask (max IoU vs all valid targets) + last-wins scatter match
    float maxiou = 0.0f;
    int hit = -1;
    for (int t = 0; t < TTx; ++t) {
      const float* e = &tab[t * 12];
      if (e[4] > 0.5f) {
        float io = iou_cwh(px, py, pw, ph, e[0], e[1], e[2], e[3]);
        maxiou = fmaxf(maxiou, io);
        if ((int)e[5] == idx) hit = t;  // last valid target wins
      }
    }
    float conf_mask = (maxiou > 0.6f) ? 0.0f : 1.0f;  // NOOBJECT_SCALE = 1
    float tconf = 0.0f;

    if (hit >= 0) {
      conf_mask = 5.0f;  // OBJECT_SCALE
      const float* e = &tab[hit * 12];
      tconf = iou_cwh(px, py, pw, ph, e[0], e[1], e[2], e[3]);
      // coord smooth-L1, COORD_SCALE/2 = 0.5
      float d, cl = 0.0f;
      d = fabsf(x - e[6]);  cl += (d < 1.0f) ? 0.5f * d * d : d - 0.5f;
      d = fabsf(y - e[7]);  cl += (d < 1.0f) ? 0.5f * d * d : d - 0.5f;
      d = fabsf(o2 - e[8]); cl += (d < 1.0f) ? 0.5f * d * d : d - 0.5f;
      d = fabsf(o3 - e[9]); cl += (d < 1.0f) ? 0.5f * d * d : d - 0.5f;
      s_main += 0.5f * cl;
      s_n = 1.0f;

      // focal class loss: softmax over first 14, sigmoid over remaining 66
      float lg[NPOSE];
      float mx = -3.0e38f;
#pragma unroll
      for (int c = 0; c < NPOSE; ++c) {
        lg[c] = base[(5 + c) * NHx * NWx];
        mx = fmaxf(mx, lg[c]);
      }
      float ssum = 0.0f;
#pragma unroll
      for (int c = 0; c < NPOSE; ++c) {
        lg[c] = expf(lg[c] - mx);
        ssum += lg[c];
      }
      const float inv = 1.0f / ssum;
      const float* trow = tcls + ((size_t)b * TTx + hit) * NCx;
      float cs = 0.0f;
      for (int c = 0; c < NCx; ++c) {
        float p;
        if (c < NPOSE) {
          p = lg[c] * inv;
        } else {
          p = sigmoidf_(base[(5 + c) * NHx * NWx]);
        }
        p = fminf(fmaxf(p, EPSC), 1.0f - EPSC);
        if (trow[c] == 1.0f)
          cs += sqrtf(1.0f - p) * logf(p) * wpos_s[c];
        else
          cs += sqrtf(p) * logf(1.0f - p) * wneg_s[c];
      }
      s_cls = cs;
    }

    const float dc = conf - tconf;
    s_main += 0.5f * conf_mask * dc * dc;  // (conf*cm - tconf*cm)^2 / 2
  }

  // ---- uniform reduction: per-wave WMMA sums, then LDS, then f64 atomics ----
  const float r0 = wave_sum_wmma(s_main);
  const float r1 = wave_sum_wmma(s_cls);
  const float r2 = wave_sum_wmma(s_n);
  const int wave = tid >> 5;
  if ((tid & 31) == 0) {
    wred[wave][0] = r0;
    wred[wave][1] = r1;
    wred[wave][2] = r2;
  }
  __syncthreads();
  if (tid == 0) {
    float a0 = 0.0f, a1 = 0.0f, a2 = 0.0f;
#pragma unroll
    for (int wv = 0; wv < 8; ++wv) {
      a0 += wred[wv][0];
      a1 += wred[wv][1];
      a2 += wred[wv][2];
    }
    atomicAdd(&acc[0], (double)a0);
    atomicAdd(&acc[1], (double)a1);
    atomicAdd(&acc[2], (double)a2);
  }
}

__global__ void fin_kernel(const double* __restrict__ acc,
                           float* __restrict__ out) {
  double n = acc[2] < 1.0 ? 1.0 : acc[2];
  out[0] = (float)(acc[0] - acc[1] / n);  // CLASS_SCALE = 1
}

extern "C" void kernel_launch(void* const* d_in, const int* in_sizes, int n_in,
                              void* d_out, int out_size, void* d_ws,
                              size_t ws_size, hipStream_t stream) {
  (void)n_in; (void)out_size; (void)ws_size;
  const float* outp = (const float*)d_in[0];
  const float* tb   = (const float*)d_in[1];
  const float* tcls = (const float*)d_in[2];
  const float* cwt  = (const float*)d_in[3];

  const int nB = in_sizes[1] / (TTx * 4);   // target_boxes is (nB, 50, 4)
  const int nT = nB * TTx;

  double* acc = (double*)d_ws;                       // 3 f64 accumulators
  float* tabg = (float*)((char*)d_ws + 64);          // nT x 12 target records

  init_kernel<<<1, 32, 0, stream>>>(acc);
  targets_kernel<<<(nT + 255) / 256, 256, 0, stream>>>(tb, nT, tabg);
  dim3 grid((CELLS + 255) / 256, nB);
  cells_kernel<<<grid, 256, 0, stream>>>(outp, tcls, cwt, tabg, acc);
  fin_kernel<<<1, 1, 0, stream>>>(acc, (float*)d_out);
}